// PCENLayer_12360915878270
// MI455X (gfx1250) — compile-verified
//
#include <hip/hip_runtime.h>
#include <cstdint>
#include <cstddef>

#define FLOOR_EPS 1e-6f
#define T_LEN     8000
#define CHUNK     32
#define NTH       256
#define NWAVE     (NTH / 32)
#define NCHUNK    ((T_LEN + CHUNK - 1) / CHUNK)   // 250 chunks per row
#define CH_STRIDE 36                              // dwords per chunk in LDS: 32 data + 4 pad (16B aligned)
#define LDS_DW    (NCHUNK * CH_STRIDE)            // 9000 dwords = 36000 bytes

typedef float v4f __attribute__((ext_vector_type(4)));   // native vector for NT stores

// ---- hardware transcendentals (v_log_f32 / v_exp_f32) ----
__device__ __forceinline__ float fast_log2(float v) { return __builtin_amdgcn_logf(v); }
__device__ __forceinline__ float fast_exp2(float v) { return __builtin_amdgcn_exp2f(v); }

// ---- CDNA5 async global -> LDS copy (gfx1250), inline asm (portable across toolchains) ----
__device__ __forceinline__ void async_ld_b128(unsigned lds_byte_addr, const float* g) {
    asm volatile("global_load_async_to_lds_b128 %0, %1, off"
                 :: "v"(lds_byte_addr), "v"((unsigned long long)(uintptr_t)g)
                 : "memory");
}
__device__ __forceinline__ void wait_async0() {
    asm volatile("s_wait_asynccnt 0" ::: "memory");
}

__global__ __launch_bounds__(NTH) void pcen_scan_kernel(
    const float* __restrict__ x,
    const float* __restrict__ alpha,
    const float* __restrict__ delta,
    const float* __restrict__ root,
    const float* __restrict__ wts,
    float* __restrict__ out,
    int C)
{
    __shared__ float s_buf[LDS_DW];
    __shared__ float sWA[NWAVE];
    __shared__ float sWB[NWAVE];

    const int rid  = blockIdx.x;          // one workgroup per (b, c) row
    const int tid  = threadIdx.x;
    const int lane = tid & 31;
    const int wave = tid >> 5;
    // channel index: fast path for power-of-two C (C == 64 here)
    const int c = ((C & (C - 1)) == 0) ? (rid & (C - 1)) : (rid % C);

    // per-channel parameters (uniform across block)
    const float a   = fminf(alpha[c], 1.0f);
    const float r   = 1.0f / fmaxf(root[c], 1.0f);
    const float d   = delta[c];
    const float w   = fminf(fmaxf(wts[c], 0.0f), 1.0f);
    const float k   = 1.0f - w;
    const float dpr = fast_exp2(r * fast_log2(d));   // d ** (1/root)

    const float* rowx = x   + (size_t)rid * T_LEN;
    float*       rowo = out + (size_t)rid * T_LEN;

    // LDS byte address of s_buf (flat shared addr low 32 bits == LDS offset)
    const unsigned lds_base = (unsigned)(size_t)&s_buf[0];

    // ---- Phase 0: coalesced async copy of the whole row into LDS (padded layout) ----
    #pragma unroll
    for (int it = 0; it < (T_LEN / 4 + NTH - 1) / NTH; ++it) {
        int v4 = it * NTH + tid;                  // float4 index within row
        if (v4 < T_LEN / 4) {
            int f = v4 * 4;                       // flat float index (multiple of 4)
            unsigned dw = (unsigned)((f >> 5) * CH_STRIDE + (f & 31));
            async_ld_b128(lds_base + dw * 4u, rowx + f);
        }
    }
    wait_async0();
    __syncthreads();

    // ---- Phase 1: per-chunk recurrence summary: carry_out = A * carry_in + B ----
    float A = 1.0f, Bv = 0.0f;
    float xa[CHUNK];
    const bool act = (tid < NCHUNK);              // threads 250..255 are identity
    if (act) {
        const float4* lv = (const float4*)(s_buf + tid * CH_STRIDE);
        #pragma unroll
        for (int j = 0; j < CHUNK / 4; ++j) {
            float4 v = lv[j];
            xa[4 * j + 0] = v.x; xa[4 * j + 1] = v.y;
            xa[4 * j + 2] = v.z; xa[4 * j + 3] = v.w;
        }
        #pragma unroll
        for (int j = 0; j < CHUNK; ++j) {
            Bv = fmaf(k, Bv, w * xa[j]);          // acc = w*x + (1-w)*acc, carry_in = 0
        }
        // A = k^CHUNK by repeated squaring (CHUNK = 32)
        float k2 = k * k, k4 = k2 * k2, k8 = k4 * k4, k16 = k8 * k8;
        A = k16 * k16;
    }

    // ---- inclusive wave32 scan of (A, B) under composition ----
    #pragma unroll
    for (int off = 1; off < 32; off <<= 1) {
        float Au = __shfl_up(A, off, 32);
        float Bu = __shfl_up(Bv, off, 32);
        if (lane >= off) { Bv = fmaf(A, Bu, Bv); A *= Au; }
    }
    if (lane == 31) { sWA[wave] = A; sWB[wave] = Bv; }   // wave totals
    // exclusive-within-wave prefix for this lane
    float Ae = __shfl_up(A, 1, 32);
    float Be = __shfl_up(Bv, 1, 32);
    if (lane == 0) { Ae = 1.0f; Be = 0.0f; }
    __syncthreads();

    // serial scan across the 8 wave totals -> exclusive per-wave prefix
    if (tid == 0) {
        float Ar = 1.0f, Br = 0.0f;
        #pragma unroll
        for (int wv = 0; wv < NWAVE; ++wv) {
            float Aw = sWA[wv], Bw = sWB[wv];
            sWA[wv] = Ar; sWB[wv] = Br;
            Br = fmaf(Aw, Br, Bw);
            Ar *= Aw;
        }
    }
    __syncthreads();

    const float init = s_buf[0];                   // ema init = x[b, c, 0]
    const float Awe  = sWA[wave];
    const float Bwe  = sWB[wave];
    // carry into this thread's chunk: lane-exclusive applied after wave-exclusive
    float carry = fmaf(Ae, fmaf(Awe, init, Bwe), Be);

    // ---- Phase 2: re-run recurrence with true carry + pointwise PCEN, store ----
    if (act) {
        float acc = carry;
        #pragma unroll
        for (int j = 0; j < CHUNK; ++j) {
            acc = fmaf(k, acc, w * xa[j]);                                // ema_t
            float g = xa[j] * fast_exp2(-a * fast_log2(acc + FLOOR_EPS)); // x / (eps+ema)^a
            xa[j] = fast_exp2(r * fast_log2(g + d)) - dpr;                // (g+d)^(1/root) - d^(1/root)
        }
        v4f* ov = (v4f*)(rowo + tid * CHUNK);
        #pragma unroll
        for (int j = 0; j < CHUNK / 4; ++j) {
            v4f v;
            v.x = xa[4 * j + 0]; v.y = xa[4 * j + 1];
            v.z = xa[4 * j + 2]; v.w = xa[4 * j + 3];
            __builtin_nontemporal_store(v, &ov[j]);   // output never re-read: NT hint
        }
    }
}

extern "C" void kernel_launch(void* const* d_in, const int* in_sizes, int n_in,
                              void* d_out, int out_size, void* d_ws, size_t ws_size,
                              hipStream_t stream) {
    const float* x     = (const float*)d_in[0];
    const float* alpha = (const float*)d_in[1];
    const float* delta = (const float*)d_in[2];
    const float* root  = (const float*)d_in[3];
    const float* wts   = (const float*)d_in[4];
    float* out = (float*)d_out;

    const int C    = in_sizes[1];                 // 64 channels
    const int rows = in_sizes[0] / T_LEN;         // B * C = 8192 rows

    pcen_scan_kernel<<<rows, NTH, 0, stream>>>(x, alpha, delta, root, wts, out, C);
}